// RBP_Veri_27058293964897
// MI455X (gfx1250) — compile-verified
//
#include <hip/hip_runtime.h>
#include <hip/hip_bf16.h>
#include <math.h>

typedef float v2f __attribute__((ext_vector_type(2)));
typedef float v8f __attribute__((ext_vector_type(8)));

#define NPAIR 128
#define HS    712
#define CB    1424   // 2*HS
#define RELN  1936
#define ACN   157
#define BIM   16

static __device__ __forceinline__ v2f ld2(const float* p) {
    return *reinterpret_cast<const v2f*>(p);
}

// ---------------------------------------------------------------------------
// Generic WMMA f32 GEMM: C[m*ldc+n] = sum_k A[m*lda+k]*B[n*ldb+k] + bias[n]
// One wave per 16x16 tile; requires K % 4 == 0, lda/ldb even (8B alignment).
// ---------------------------------------------------------------------------
__global__ void gemm_wmma_f32(const float* __restrict__ A, int lda,
                              const float* __restrict__ B, int ldb,
                              const float* __restrict__ bias,
                              float* __restrict__ C, int ldc,
                              int M, int N, int K) {
    const int wave = threadIdx.x >> 5;
    const int lane = threadIdx.x & 31;
    const int nt = blockIdx.x * 8 + wave;
    const int mt = blockIdx.y;
    const int nTiles = (N + 15) >> 4;
    if (nt >= nTiles) return;

    const int half = lane >> 4;   // 0: K{0,1}, 1: K{2,3}
    const int l16  = lane & 15;

    int arow = mt * 16 + l16; if (arow >= M) arow = M - 1;
    int bcol = nt * 16 + l16; if (bcol >= N) bcol = N - 1;
    const float* pa = A + (size_t)arow * lda + 2 * half;
    const float* pb = B + (size_t)bcol * ldb + 2 * half;

    v8f c = {};
    #pragma unroll 4
    for (int k = 0; k < K; k += 4) {
        v2f a = ld2(pa + k);
        v2f b = ld2(pb + k);
        c = __builtin_amdgcn_wmma_f32_16x16x4_f32(false, a, false, b,
                                                  (short)0, c, false, false);
    }

    const int n = nt * 16 + l16;
    if (n < N) {
        float bv = bias ? bias[n] : 0.0f;
        #pragma unroll
        for (int r = 0; r < 8; ++r) {
            int m = mt * 16 + r + 8 * half;
            if (m < M) C[(size_t)m * ldc + n] = c[r] + bv;
        }
    }
}

// ---------------------------------------------------------------------------
// Fused bilinear: bil[n,o] = sum_i subj[n,i] * (W[o] @ obj[n,:])[i] + bil_b[o]
// comb = [subj | obj] rows of length 1424. One block per o; wave w owns
// n-tile w; loops i-tiles of Q = W[o] (i x j) @ obj^T (j x n) via WMMA and
// folds subj into a scalar accumulator per lane. Writes zcat[:, 712+o].
// ---------------------------------------------------------------------------
__global__ void bilinear_wmma(const float* __restrict__ W,     // (712,712,712)
                              const float* __restrict__ comb,  // (128,1424)
                              const float* __restrict__ bil_b, // (712,)
                              float* __restrict__ zcat) {      // (128,1424)
    const int o = blockIdx.x;
    const int wave = threadIdx.x >> 5;
    const int lane = threadIdx.x & 31;
    const int half = lane >> 4;
    const int l16  = lane & 15;
    const int n = wave * 16 + l16;            // 0..127 exactly

    const float* Wo = W + (size_t)o * (HS * HS);
    const float* pb = comb + (size_t)n * CB + HS + 2 * half;   // obj half
    const float* subj_row = comb + (size_t)n * CB;             // subj half

    float sum = 0.0f;
    for (int t = 0; t < 45; ++t) {
        const int i0 = t * 16;
        int irow = i0 + l16; if (irow >= HS) irow = HS - 1;
        const float* pa = Wo + (size_t)irow * HS + 2 * half;
        v8f c = {};
        #pragma unroll 4
        for (int k = 0; k < HS; k += 4) {
            v2f a = ld2(pa + k);
            v2f b = ld2(pb + k);
            c = __builtin_amdgcn_wmma_f32_16x16x4_f32(false, a, false, b,
                                                      (short)0, c, false, false);
        }
        #pragma unroll
        for (int r = 0; r < 8; ++r) {
            int iv = i0 + r + 8 * half;
            float s = (iv < HS) ? subj_row[iv] : 0.0f;
            sum += c[r] * s;
        }
    }
    sum += __shfl_xor(sum, 16, 32);           // combine the two half-lanes (same n)
    if (half == 0) zcat[(size_t)n * CB + HS + o] = sum + bil_b[o];
}

// ---------------------------------------------------------------------------
// Glue kernels
// ---------------------------------------------------------------------------
__global__ void gather_feats(const float* __restrict__ features,
                             const int* __restrict__ pair_idx,
                             float* __restrict__ featS, float* __restrict__ featO) {
    int idx = blockIdx.x * blockDim.x + threadIdx.x;
    if (idx >= NPAIR * 2048) return;
    int nn = idx >> 11, k = idx & 2047;
    featS[idx] = features[(size_t)pair_idx[2 * nn]     * 2048 + k];
    featO[idx] = features[(size_t)pair_idx[2 * nn + 1] * 2048 + k];
}

__global__ void fill_emb(const int* __restrict__ labels,
                         const int* __restrict__ pair_idx,
                         const float* __restrict__ emb1,
                         const float* __restrict__ emb2,
                         float* __restrict__ comb) {
    int idx = blockIdx.x * blockDim.x + threadIdx.x;
    if (idx >= NPAIR * 200) return;
    int nn = idx / 200, t = idx % 200;
    int ls = labels[pair_idx[2 * nn]];
    int lo = labels[pair_idx[2 * nn + 1]];
    comb[(size_t)nn * CB + 512 + t]        = emb1[ls * 200 + t];
    comb[(size_t)nn * CB + HS + 512 + t]   = emb2[lo * 200 + t];
}

__global__ void conv_union(const float* __restrict__ uf, const float* __restrict__ w,
                           const float* __restrict__ b, float* __restrict__ upm) {
    int idx = blockIdx.x * blockDim.x + threadIdx.x;
    if (idx >= NPAIR * 256 * 49) return;
    int np = idx / (256 * 49), r = idx % (256 * 49);
    int oc = r / 49, p = r % 49;
    const float* a = uf + (size_t)np * 1024 * 49 + p;
    const float* ww = w + (size_t)oc * 1024;
    float s = b[oc];
    for (int c = 0; c < 1024; ++c) s += a[c * 49] * ww[c];
    upm[idx] = s;
}

__global__ void conv1_bn(const float* __restrict__ x, const float* __restrict__ w,
                         const float* __restrict__ cb,
                         const float* __restrict__ bs, const float* __restrict__ bb,
                         const float* __restrict__ bm, const float* __restrict__ bv,
                         float* __restrict__ m1) {
    int idx = blockIdx.x * blockDim.x + threadIdx.x;
    if (idx >= NPAIR * 128 * 196) return;
    int np = idx / (128 * 196), r = idx % (128 * 196);
    int oc = r / 196, r2 = r % 196, y = r2 / 14, xx = r2 % 14;
    float s = cb[oc];
    for (int ic = 0; ic < 2; ++ic)
        #pragma unroll
        for (int ky = 0; ky < 7; ++ky) {
            int iy = 2 * y - 3 + ky;
            if (iy < 0 || iy >= 27) continue;
            #pragma unroll
            for (int kx = 0; kx < 7; ++kx) {
                int ix = 2 * xx - 3 + kx;
                if (ix < 0 || ix >= 27) continue;
                s += x[((size_t)np * 2 + ic) * 729 + iy * 27 + ix] *
                     w[((oc * 2 + ic) * 7 + ky) * 7 + kx];
            }
        }
    float v = fmaxf(s, 0.0f);
    m1[idx] = (v - bm[oc]) * (bs[oc] * rsqrtf(bv[oc] + 1e-5f)) + bb[oc];
}

__global__ void maxpool3s2(const float* __restrict__ m1, float* __restrict__ mp) {
    int idx = blockIdx.x * blockDim.x + threadIdx.x;
    if (idx >= NPAIR * 128 * 49) return;
    int np = idx / (128 * 49), r = idx % (128 * 49);
    int c = r / 49, p = r % 49, y = p / 7, xx = p % 7;
    float best = -INFINITY;
    #pragma unroll
    for (int ky = 0; ky < 3; ++ky) {
        int iy = 2 * y - 1 + ky;
        if (iy < 0 || iy >= 14) continue;
        #pragma unroll
        for (int kx = 0; kx < 3; ++kx) {
            int ix = 2 * xx - 1 + kx;
            if (ix < 0 || ix >= 14) continue;
            best = fmaxf(best, m1[((size_t)np * 128 + c) * 196 + iy * 14 + ix]);
        }
    }
    mp[idx] = best;
}

__global__ void conv2_bn_add(const float* __restrict__ mp, const float* __restrict__ w,
                             const float* __restrict__ cb,
                             const float* __restrict__ bs, const float* __restrict__ bb,
                             const float* __restrict__ bm, const float* __restrict__ bv,
                             float* __restrict__ upm) {
    int idx = blockIdx.x * blockDim.x + threadIdx.x;
    if (idx >= NPAIR * 256 * 49) return;
    int np = idx / (256 * 49), r = idx % (256 * 49);
    int oc = r / 49, p = r % 49, y = p / 7, xx = p % 7;
    float s = cb[oc];
    for (int ic = 0; ic < 128; ++ic) {
        const float* in = mp + ((size_t)np * 128 + ic) * 49;
        const float* ww = w + ((size_t)oc * 128 + ic) * 9;
        #pragma unroll
        for (int ky = 0; ky < 3; ++ky) {
            int iy = y - 1 + ky;
            if (iy < 0 || iy >= 7) continue;
            #pragma unroll
            for (int kx = 0; kx < 3; ++kx) {
                int ix = xx - 1 + kx;
                if (ix < 0 || ix >= 7) continue;
                s += in[iy * 7 + ix] * ww[ky * 3 + kx];
            }
        }
    }
    float v = fmaxf(s, 0.0f);
    upm[idx] += (v - bm[oc]) * (bs[oc] * rsqrtf(bv[oc] + 1e-5f)) + bb[oc];
}

__global__ void relu_inplace(float* __restrict__ x, int n) {
    int idx = blockIdx.x * blockDim.x + threadIdx.x;
    if (idx < n) x[idx] = fmaxf(x[idx], 0.0f);
}

__global__ void seg_max(const float* __restrict__ rel, const int* __restrict__ im_idx,
                        float* __restrict__ pooled) {
    int idx = blockIdx.x * blockDim.x + threadIdx.x;
    if (idx >= BIM * RELN) return;
    int b = idx / RELN, k = idx % RELN;
    float mx = -INFINITY;
    for (int nn = 0; nn < NPAIR; ++nn)
        if (im_idx[nn] == b) mx = fmaxf(mx, rel[(size_t)nn * RELN + k]);
    pooled[idx] = mx;
}

__global__ void final_logits(const float* __restrict__ pA, const float* __restrict__ pB,
                             const float* __restrict__ ac2_w, const float* __restrict__ ac2_b,
                             float* __restrict__ out) {
    int idx = blockIdx.x * blockDim.x + threadIdx.x;
    if (idx >= BIM * ACN) return;
    int b = idx / ACN, a = idx % ACN;
    const float* ra = pA + (size_t)b * RELN;
    const float* rb = pB + (size_t)a * RELN;
    float s = ac2_b[0];
    for (int k = 0; k < RELN; ++k)
        s += fmaxf(ra[k] + rb[k], 0.0f) * ac2_w[k];
    out[idx] = 1.0f / (1.0f + expf(-s));
}

// ---------------------------------------------------------------------------
static inline int cdiv(int a, int b) { return (a + b - 1) / b; }

extern "C" void kernel_launch(void* const* d_in, const int* in_sizes, int n_in,
                              void* d_out, int out_size, void* d_ws, size_t ws_size,
                              hipStream_t stream) {
    const float* features = (const float*)d_in[0];
    const int*   labels   = (const int*)d_in[1];
    const int*   im_idx   = (const int*)d_in[2];
    const int*   pair_idx = (const int*)d_in[3];
    const float* union_feat = (const float*)d_in[4];
    const float* spatial  = (const float*)d_in[5];
    const float* class_enc= (const float*)d_in[6];
    const float* W_union  = (const float*)d_in[7];
    const float* b_union  = (const float*)d_in[8];
    const float* conv1_w  = (const float*)d_in[9];
    const float* conv1_b  = (const float*)d_in[10];
    const float* bn1_s = (const float*)d_in[11], *bn1_b = (const float*)d_in[12];
    const float* bn1_m = (const float*)d_in[13], *bn1_v = (const float*)d_in[14];
    const float* conv2_w  = (const float*)d_in[15];
    const float* conv2_b  = (const float*)d_in[16];
    const float* bn2_s = (const float*)d_in[17], *bn2_b = (const float*)d_in[18];
    const float* bn2_m = (const float*)d_in[19], *bn2_v = (const float*)d_in[20];
    const float* subj_w = (const float*)d_in[21], *subj_b = (const float*)d_in[22];
    const float* obj_w  = (const float*)d_in[23], *obj_b  = (const float*)d_in[24];
    const float* vr_w   = (const float*)d_in[25], *vr_b   = (const float*)d_in[26];
    const float* emb1   = (const float*)d_in[27], *emb2   = (const float*)d_in[28];
    const float* bil_w  = (const float*)d_in[29], *bil_b  = (const float*)d_in[30];
    const float* lin_w  = (const float*)d_in[31], *lin_b  = (const float*)d_in[32];
    const float* proj_w = (const float*)d_in[33], *proj_b = (const float*)d_in[34];
    const float* ac1_w  = (const float*)d_in[35], *ac1_b  = (const float*)d_in[36];
    const float* ac2_w  = (const float*)d_in[37], *ac2_b  = (const float*)d_in[38];
    float* out = (float*)d_out;

    float* ws = (float*)d_ws;
    size_t off = 0;
    float* featS  = ws + off; off += (size_t)NPAIR * 2048;
    float* featO  = ws + off; off += (size_t)NPAIR * 2048;
    float* comb   = ws + off; off += (size_t)NPAIR * CB;
    float* upm    = ws + off; off += (size_t)NPAIR * 256 * 49;
    float* m1     = ws + off; off += (size_t)NPAIR * 128 * 196;
    float* mp     = ws + off; off += (size_t)NPAIR * 128 * 49;
    float* zcat   = ws + off; off += (size_t)NPAIR * CB;
    float* rel    = ws + off; off += (size_t)NPAIR * RELN;
    float* pooled = ws + off; off += (size_t)BIM * RELN;
    float* pA     = ws + off; off += (size_t)BIM * RELN;
    float* pB     = ws + off; off += (size_t)ACN * RELN;

    const int T = 256;

    // 1) gather features for subject/object
    gather_feats<<<cdiv(NPAIR * 2048, T), T, 0, stream>>>(features, pair_idx, featS, featO);
    // 2-3) subj/obj projections -> comb[:, :512] and comb[:, 712:1224]
    gemm_wmma_f32<<<dim3(4, 8), T, 0, stream>>>(featS, 2048, subj_w, 2048, subj_b,
                                                comb, CB, NPAIR, 512, 2048);
    gemm_wmma_f32<<<dim3(4, 8), T, 0, stream>>>(featO, 2048, obj_w, 2048, obj_b,
                                                comb + HS, CB, NPAIR, 512, 2048);
    // 4) embedding fills -> comb[:, 512:712] and comb[:, 1224:1424]
    fill_emb<<<cdiv(NPAIR * 200, T), T, 0, stream>>>(labels, pair_idx, emb1, emb2, comb);
    // 5) u = 1x1 conv -> upm
    conv_union<<<cdiv(NPAIR * 256 * 49, T), T, 0, stream>>>(union_feat, W_union, b_union, upm);
    // 6-8) mask branch: conv1 -> relu -> bn -> maxpool -> conv2 -> relu -> bn, add into upm
    conv1_bn<<<cdiv(NPAIR * 128 * 196, T), T, 0, stream>>>(spatial, conv1_w, conv1_b,
                                                           bn1_s, bn1_b, bn1_m, bn1_v, m1);
    maxpool3s2<<<cdiv(NPAIR * 128 * 49, T), T, 0, stream>>>(m1, mp);
    conv2_bn_add<<<cdiv(NPAIR * 256 * 49, T), T, 0, stream>>>(mp, conv2_w, conv2_b,
                                                              bn2_s, bn2_b, bn2_m, bn2_v, upm);
    // 9) vr = (u+m) @ vr_w.T + vr_b  -> rel[:, 1424:1936]
    gemm_wmma_f32<<<dim3(4, 8), T, 0, stream>>>(upm, 12544, vr_w, 12544, vr_b,
                                                rel + CB, RELN, NPAIR, 512, 12544);
    // 10) bilinear -> zcat[:, 712:1424]   (the 92-GFLOP hot loop)
    bilinear_wmma<<<dim3(HS), T, 0, stream>>>(bil_w, comb, bil_b, zcat);
    // 11) lin -> zcat[:, :712]
    gemm_wmma_f32<<<dim3(6, 8), T, 0, stream>>>(comb, CB, lin_w, CB, lin_b,
                                                zcat, CB, NPAIR, HS, CB);
    // 12) relu over zcat
    relu_inplace<<<cdiv(NPAIR * CB, T), T, 0, stream>>>(zcat, NPAIR * CB);
    // 13) proj -> rel[:, :1424]
    gemm_wmma_f32<<<dim3(12, 8), T, 0, stream>>>(zcat, CB, proj_w, CB, proj_b,
                                                 rel, RELN, NPAIR, CB, CB);
    // 14) segment max over pairs -> pooled
    seg_max<<<cdiv(BIM * RELN, T), T, 0, stream>>>(rel, im_idx, pooled);
    // 15) pA = pooled @ ac1_w[:, :1936].T + ac1_b
    gemm_wmma_f32<<<dim3(16, 1), T, 0, stream>>>(pooled, RELN, ac1_w, 2448, ac1_b,
                                                 pA, RELN, BIM, RELN, RELN);
    // 16) pB = class_enc @ ac1_w[:, 1936:].T
    gemm_wmma_f32<<<dim3(16, 10), T, 0, stream>>>(class_enc, 512, ac1_w + RELN, 2448,
                                                  (const float*)nullptr,
                                                  pB, RELN, ACN, RELN, 512);
    // 17) logits + sigmoid
    final_logits<<<cdiv(BIM * ACN, T), T, 0, stream>>>(pA, pB, ac2_w, ac2_b, out);
}